// SobelModule_60748017434829
// MI455X (gfx1250) — compile-verified
//
#include <hip/hip_runtime.h>
#include <stdint.h>

// Problem constants from the reference: B=16, C=64, H=W=128, f32.
#define H       128
#define W       128
#define PLANE   (H * W)          // 16384 floats per channel plane
#define TILE_H  16               // output rows per block
#define LDS_W   132              // 130 used cols (+2 pad), keeps float4 alignment
#define LDS_R   18               // TILE_H + 2 halo rows

__global__ __launch_bounds__(256) void sobel_async_kernel(
    const float* __restrict__ f1,
    const float* __restrict__ f2,
    float* __restrict__ out,
    size_t per_tensor)           // elements per output tensor (B*C*H*W)
{
    __shared__ __align__(16) float tile[LDS_R * LDS_W];

    const int tid   = threadIdx.x;
    const int t     = blockIdx.x;          // row tile 0..7
    const int plane = blockIdx.y;          // 0..B*C-1
    const int f     = blockIdx.z;          // which input
    const int r0    = t * TILE_H;

    const float* src = (f == 0 ? f1 : f2) + (size_t)plane * PLANE;

    // ---- Zero only the halo cells the async copies will NOT write.
    //      (Disjoint addresses -> immune to DS/async ordering rules.)
    if (tid < LDS_R)                    tile[tid * LDS_W + 0]          = 0.0f; // col -1
    if (tid >= 32 && tid < 32 + LDS_R)  tile[(tid - 32) * LDS_W + 129] = 0.0f; // col 128
    if (r0 == 0 && tid < W)             tile[0 * LDS_W + 1 + tid]      = 0.0f; // top halo row
    if (r0 == H - TILE_H && tid < W)    tile[(LDS_R - 1) * LDS_W + 1 + tid] = 0.0f; // bottom halo

    // ---- Async copy interior: rows r0-1 .. r0+16 (clipped), 128 cols each.
    //      18*128 = 2304 words, 9 per thread. Tracked on ASYNCcnt, no VGPR data.
    #pragma unroll
    for (int k = 0; k < 9; ++k) {
        int idx  = tid + k * 256;          // 0 .. 2303
        int row  = idx >> 7;               // 0 .. 17 (LDS row)
        int col  = idx & 127;              // 0 .. 127 (global col)
        int grow = r0 - 1 + row;           // global row
        if (grow >= 0 && grow < H) {
            // Generic LDS pointer low 32 bits == LDS byte offset (aperture layout).
            unsigned lds_off = (unsigned)(uintptr_t)&tile[row * LDS_W + col + 1];
            unsigned g_off   = (unsigned)((grow << 7) + col) * 4u;
            asm volatile("global_load_async_to_lds_b32 %0, %1, %2"
                         :
                         : "v"(lds_off), "v"(g_off), "s"(src)
                         : "memory");
        }
    }
    asm volatile("s_wait_asynccnt 0" ::: "memory");
    __syncthreads();

    // ---- Compute: each thread -> 2 rows x 4 cols, from a 4x6 LDS window.
    const int c   = (tid & 31) * 4;        // output col base 0..124
    const int rg  = tid >> 5;              // 0..7
    const int rl0 = rg * 2;                // local output row base

    float R[4][6];
    #pragma unroll
    for (int rr = 0; rr < 4; ++rr) {
        const float* p = &tile[(rl0 + rr) * LDS_W + c];   // 16B aligned
        float4 v4 = *(const float4*)p;                    // ds_load_b128
        float2 v2 = *(const float2*)(p + 4);              // ds_load_b64
        R[rr][0] = v4.x; R[rr][1] = v4.y; R[rr][2] = v4.z;
        R[rr][3] = v4.w; R[rr][4] = v2.x; R[rr][5] = v2.y;
    }

    float* outx = out + (size_t)(2 * f + 0) * per_tensor + (size_t)plane * PLANE;
    float* outy = out + (size_t)(2 * f + 1) * per_tensor + (size_t)plane * PLANE;

    #pragma unroll
    for (int j = 0; j < 2; ++j) {
        float4 sx, sy;
        float* px = (float*)&sx;
        float* py = (float*)&sy;
        #pragma unroll
        for (int i = 0; i < 4; ++i) {
            // window: R[j] = row above, R[j+1] = center row, R[j+2] = row below
            float tl = R[j][i],     tm = R[j][i + 1],     tr = R[j][i + 2];
            float ml = R[j + 1][i],                       mr = R[j + 1][i + 2];
            float bl = R[j + 2][i], bm = R[j + 2][i + 1], br = R[j + 2][i + 2];
            // SOBEL_X = [[1,0,-1],[2,0,-2],[1,0,-1]]
            px[i] = (tl - tr) + 2.0f * (ml - mr) + (bl - br);
            // SOBEL_Y = [[1,2,1],[0,0,0],[-1,-2,-1]]
            py[i] = (tl + 2.0f * tm + tr) - (bl + 2.0f * bm + br);
        }
        size_t o = (size_t)(r0 + rl0 + j) * W + c;        // 16B aligned
        *(float4*)(outx + o) = sx;                        // global_store_b128
        *(float4*)(outy + o) = sy;                        // global_store_b128
    }
}

extern "C" void kernel_launch(void* const* d_in, const int* in_sizes, int n_in,
                              void* d_out, int out_size, void* d_ws, size_t ws_size,
                              hipStream_t stream) {
    (void)n_in; (void)out_size; (void)d_ws; (void)ws_size;
    const float* f1 = (const float*)d_in[0];
    const float* f2 = (const float*)d_in[1];
    float* out = (float*)d_out;

    const size_t per_tensor = (size_t)in_sizes[0];        // B*C*H*W
    const unsigned planes   = (unsigned)(per_tensor / PLANE);

    dim3 grid(H / TILE_H, planes, 2);                     // (8, 1024, 2)
    sobel_async_kernel<<<grid, 256, 0, stream>>>(f1, f2, out, per_tensor);
}